// GGNN1_74569222193909
// MI455X (gfx1250) — compile-verified
//
#include <hip/hip_runtime.h>
#include <hip/hip_bf16.h>

#define N_NODES  50000
#define N_EDGES  800000
#define D1       64
#define D2       32
#define N_CLASSES 6
#define N_GRAPHS 128
#define M_TILES  (N_NODES / 16)   // 3125, exact

typedef __attribute__((ext_vector_type(2))) float v2f;
typedef __attribute__((ext_vector_type(8))) float v8f;

__device__ __forceinline__ float sigm(float x) { return 1.0f / (1.0f + __expf(-x)); }
__device__ __forceinline__ float tanh_fast(float x) {
    // tanh(x) = 1 - 2/(exp(2x)+1)
    return 1.0f - 2.0f / (__expf(2.0f * x) + 1.0f);
}

// ---------------------------------------------------------------------------
// zero fill
// ---------------------------------------------------------------------------
__global__ void zero_kernel(float* __restrict__ p, int n) {
    int t = blockIdx.x * blockDim.x + threadIdx.x;
    if (t < n) p[t] = 0.0f;
}

// ---------------------------------------------------------------------------
// m = H @ W   (H: [N,64], W: [64,64] row-major, M_out: [N,64])
// One wave (32 threads) computes a 16x16 tile via V_WMMA_F32_16X16X4_F32.
// grid = (3125, 4), block = 32
// ---------------------------------------------------------------------------
__global__ void gemm_hW(const float* __restrict__ H, const float* __restrict__ W,
                        float* __restrict__ M_out) {
    const int mtile = blockIdx.x;
    const int ntile = blockIdx.y;
    const int lane  = threadIdx.x;
    const int l15   = lane & 15;
    const int kb    = (lane >> 4) * 2;      // 0 or 2 (A/B K-split across half-waves)

    const float* hrow = H + (mtile * 16 + l15) * D1;
    const int    ncol = ntile * 16 + l15;

    v8f acc = {};
#pragma unroll
    for (int k = 0; k < D1; k += 4) {
        v2f a, b;
        a.x = hrow[k + kb];
        a.y = hrow[k + kb + 1];
        b.x = W[(k + kb) * D1 + ncol];
        b.y = W[(k + kb + 1) * D1 + ncol];
        acc = __builtin_amdgcn_wmma_f32_16x16x4_f32(
            false, a, false, b, (short)0, acc, false, false);
    }

    const int rbase = mtile * 16 + ((lane >> 4) << 3);   // rows r..r+7 / r+8..r+15
#pragma unroll
    for (int r = 0; r < 8; ++r)
        M_out[(rbase + r) * D1 + ncol] = acc[r];
}

// ---------------------------------------------------------------------------
// agg[dst] += m[src]   16 threads per edge, float4 gather + 4 hw fp32 atomics
// ---------------------------------------------------------------------------
__global__ void scatter_kernel(const float* __restrict__ M, const int* __restrict__ src,
                               const int* __restrict__ dst, float* __restrict__ AGG) {
    int t = blockIdx.x * blockDim.x + threadIdx.x;
    int e = t >> 4;
    if (e >= N_EDGES) return;
    int f = (t & 15) * 4;
    int s = src[e];
    int d = dst[e];
    const float4 v = *(const float4*)(M + s * D1 + f);
    float* base = AGG + d * D1 + f;
    unsafeAtomicAdd(base + 0, v.x);
    unsafeAtomicAdd(base + 1, v.y);
    unsafeAtomicAdd(base + 2, v.z);
    unsafeAtomicAdd(base + 3, v.w);
}

// ---------------------------------------------------------------------------
// GRU cell: h' = (1-z)*n + z*h with
//   gi = agg @ w_ih^T + b_ih ; gh = h @ w_hh^T + b_hh  (w_* : [192,64] row-major)
// One wave per (16-node tile, 16-col tile). Six K=64 WMMA accumulations.
// grid = (3125, 4), block = 32
// ---------------------------------------------------------------------------
__global__ void gru_kernel(const float* __restrict__ AGG, const float* __restrict__ H,
                           const float* __restrict__ w_ih, const float* __restrict__ w_hh,
                           const float* __restrict__ b_ih, const float* __restrict__ b_hh,
                           float* __restrict__ Hout) {
    const int mtile = blockIdx.x;
    const int jt    = blockIdx.y;
    const int lane  = threadIdx.x;
    const int l15   = lane & 15;
    const int kb    = (lane >> 4) * 2;
    const int ncol  = jt * 16 + l15;        // output column 0..63

    const float* arow = AGG + (mtile * 16 + l15) * D1;
    const float* hrow = H   + (mtile * 16 + l15) * D1;

    // B element for gate g, column n, depth k:  w[(g*64 + n)*64 + k]
    const float* wi_r = w_ih + (0 * D1 + ncol) * D1;
    const float* wi_z = w_ih + (1 * D1 + ncol) * D1;
    const float* wi_n = w_ih + (2 * D1 + ncol) * D1;
    const float* wh_r = w_hh + (0 * D1 + ncol) * D1;
    const float* wh_z = w_hh + (1 * D1 + ncol) * D1;
    const float* wh_n = w_hh + (2 * D1 + ncol) * D1;

    v8f ir = {}, iz = {}, in_ = {}, hr = {}, hz = {}, hn = {};
#pragma unroll
    for (int k = 0; k < D1; k += 4) {
        const int k0 = k + kb, k1 = k + kb + 1;
        v2f aa, ah, b;
        aa.x = arow[k0]; aa.y = arow[k1];
        ah.x = hrow[k0]; ah.y = hrow[k1];

        b.x = wi_r[k0]; b.y = wi_r[k1];
        ir  = __builtin_amdgcn_wmma_f32_16x16x4_f32(false, aa, false, b, (short)0, ir, false, false);
        b.x = wi_z[k0]; b.y = wi_z[k1];
        iz  = __builtin_amdgcn_wmma_f32_16x16x4_f32(false, aa, false, b, (short)0, iz, false, false);
        b.x = wi_n[k0]; b.y = wi_n[k1];
        in_ = __builtin_amdgcn_wmma_f32_16x16x4_f32(false, aa, false, b, (short)0, in_, false, false);
        b.x = wh_r[k0]; b.y = wh_r[k1];
        hr  = __builtin_amdgcn_wmma_f32_16x16x4_f32(false, ah, false, b, (short)0, hr, false, false);
        b.x = wh_z[k0]; b.y = wh_z[k1];
        hz  = __builtin_amdgcn_wmma_f32_16x16x4_f32(false, ah, false, b, (short)0, hz, false, false);
        b.x = wh_n[k0]; b.y = wh_n[k1];
        hn  = __builtin_amdgcn_wmma_f32_16x16x4_f32(false, ah, false, b, (short)0, hn, false, false);
    }

    const float bir = b_ih[ncol],        bhr = b_hh[ncol];
    const float biz = b_ih[D1 + ncol],   bhz = b_hh[D1 + ncol];
    const float bin = b_ih[2*D1 + ncol], bhn = b_hh[2*D1 + ncol];

    const int rbase = mtile * 16 + ((lane >> 4) << 3);
#pragma unroll
    for (int r = 0; r < 8; ++r) {
        const int row = rbase + r;
        const float rr = sigm((ir[r] + bir) + (hr[r] + bhr));
        const float zz = sigm((iz[r] + biz) + (hz[r] + bhz));
        const float nn = tanh_fast((in_[r] + bin) + rr * (hn[r] + bhn));
        const float hp = H[row * D1 + ncol];
        Hout[row * D1 + ncol] = (1.0f - zz) * nn + zz * hp;
    }
}

// ---------------------------------------------------------------------------
// ReLU + per-graph scatter for global mean pool
// ---------------------------------------------------------------------------
__global__ void pool_scatter(const float* __restrict__ H, const int* __restrict__ batch,
                             float* __restrict__ sums, float* __restrict__ counts) {
    int t = blockIdx.x * blockDim.x + threadIdx.x;
    int node = t >> 6;
    if (node >= N_NODES) return;
    int c = t & 63;
    float v = fmaxf(H[node * D1 + c], 0.0f);
    int g = batch[node];
    unsafeAtomicAdd(&sums[g * D1 + c], v);
    if (c == 0) unsafeAtomicAdd(&counts[g], 1.0f);
}

// ---------------------------------------------------------------------------
// pooled = sums/max(counts,1); FC1 + relu; FC2; log_softmax.  1 block x 128.
// ---------------------------------------------------------------------------
__global__ void head_kernel(const float* __restrict__ sums, const float* __restrict__ counts,
                            const float* __restrict__ fc1_w, const float* __restrict__ fc1_b,
                            const float* __restrict__ fc2_w, const float* __restrict__ fc2_b,
                            float* __restrict__ out) {
    int g = threadIdx.x;
    if (g >= N_GRAPHS) return;
    float inv = 1.0f / fmaxf(counts[g], 1.0f);
    float p[D1];
#pragma unroll
    for (int c = 0; c < D1; ++c) p[c] = sums[g * D1 + c] * inv;

    float a1[D2];
    for (int j = 0; j < D2; ++j) {
        float acc = fc1_b[j];
        const float* wr = fc1_w + j * D1;
#pragma unroll
        for (int k = 0; k < D1; ++k) acc += wr[k] * p[k];
        a1[j] = fmaxf(acc, 0.0f);
    }

    float o[N_CLASSES];
    float mx = -1e30f;
    for (int j = 0; j < N_CLASSES; ++j) {
        float acc = fc2_b[j];
        const float* wr = fc2_w + j * D2;
#pragma unroll
        for (int k = 0; k < D2; ++k) acc += wr[k] * a1[k];
        o[j] = acc;
        mx = fmaxf(mx, acc);
    }
    float se = 0.0f;
    for (int j = 0; j < N_CLASSES; ++j) se += __expf(o[j] - mx);
    float lse = mx + __logf(se);
    for (int j = 0; j < N_CLASSES; ++j) out[g * N_CLASSES + j] = o[j] - lse;
}

// ---------------------------------------------------------------------------
extern "C" void kernel_launch(void* const* d_in, const int* in_sizes, int n_in,
                              void* d_out, int out_size, void* d_ws, size_t ws_size,
                              hipStream_t stream) {
    const float* x     = (const float*)d_in[0];
    const int*   eidx  = (const int*)d_in[1];      // [2, E]
    const int*   batch = (const int*)d_in[2];
    const float* W     = (const float*)d_in[3];    // [2, 64, 64]
    const float* w_ih  = (const float*)d_in[4];    // [192, 64]
    const float* w_hh  = (const float*)d_in[5];    // [192, 64]
    const float* b_ih  = (const float*)d_in[6];
    const float* b_hh  = (const float*)d_in[7];
    const float* fc1_w = (const float*)d_in[8];
    const float* fc1_b = (const float*)d_in[9];
    const float* fc2_w = (const float*)d_in[10];
    const float* fc2_b = (const float*)d_in[11];
    float* out = (float*)d_out;

    const int* src = eidx;
    const int* dst = eidx + N_EDGES;

    // workspace layout (floats)
    const size_t NF = (size_t)N_NODES * D1;  // 3,200,000
    float* m_buf  = (float*)d_ws;
    float* agg    = m_buf + NF;
    float* h1     = agg + NF;
    float* h2     = h1 + NF;
    float* sums   = h2 + NF;                 // 128*64
    float* counts = sums + N_GRAPHS * D1;    // 128

    const dim3 tileGrid(M_TILES, 4);
    const int ZB = 256;

    for (int layer = 0; layer < 2; ++layer) {
        const float* Hcur = (layer == 0) ? x : h1;
        float*       Hout = (layer == 0) ? h1 : h2;

        zero_kernel<<<(int)((NF + ZB - 1) / ZB), ZB, 0, stream>>>(agg, (int)NF);
        gemm_hW<<<tileGrid, 32, 0, stream>>>(Hcur, W + layer * D1 * D1, m_buf);
        scatter_kernel<<<(N_EDGES * 16) / ZB, ZB, 0, stream>>>(m_buf, src, dst, agg);
        gru_kernel<<<tileGrid, 32, 0, stream>>>(agg, Hcur, w_ih, w_hh, b_ih, b_hh, Hout);
    }

    zero_kernel<<<(N_GRAPHS * D1 + N_GRAPHS + ZB - 1) / ZB, ZB, 0, stream>>>(
        sums, N_GRAPHS * D1 + N_GRAPHS);
    pool_scatter<<<(N_NODES * D1 + ZB - 1) / ZB, ZB, 0, stream>>>(h2, batch, sums, counts);
    head_kernel<<<1, 128, 0, stream>>>(sums, counts, fc1_w, fc1_b, fc2_w, fc2_b, out);
}